// SphericalHarmonicTransform_9277129359884
// MI455X (gfx1250) — compile-verified
//
#include <hip/hip_runtime.h>
#include <math.h>

// CDNA5 / gfx1250 spherical-harmonic moment reduction.
// Core contraction Mom[p][q][s] = sum_n xm^q * xp^p * w * z^s is done with
// V_WMMA_F32_16X16X4_F32 (f32 in/out: required, since powers up to degree 12
// overflow f16 and lose too much mantissa in bf16).

typedef __attribute__((ext_vector_type(2))) float v2f;
typedef __attribute__((ext_vector_type(8))) float v8f;

#define LMAX 10
#define NLM 66          // (L+1)(L+2)/2 distinct (l,m>=0) pairs
#define PART_STRIDE 132 // 2*NLM floats of partial clm per wave

__host__ __device__ constexpr double factd(int n) {
  double r = 1.0;
  for (int i = 2; i <= n; ++i) r *= (double)i;
  return r;
}

__device__ __forceinline__ v8f wmma_f32x4(v2f a, v2f b, v8f c) {
  // D = A(16x4, f32) x B(4x16, f32) + C(16x16, f32)
  return __builtin_amdgcn_wmma_f32_16x16x4_f32(
      /*neg_a=*/false, a, /*neg_b=*/false, b,
      /*c_mod=*/(short)0, c, /*reuse_a=*/false, /*reuse_b=*/false);
}

// ---------------------------------------------------------------------------
// Kernel A: each wave accumulates Mom tiles over a strided set of 4-point
// groups with f32 WMMA, then compacts to 66 complex partial clm values.
// ---------------------------------------------------------------------------
__global__ void __launch_bounds__(256, 1)
sht_partial_kernel(const float* __restrict__ pos, int N,
                   float* __restrict__ part, int NW) {
  const int lane = threadIdx.x & 31;
  const int hi   = lane >> 4;   // 0 -> K slots {0,1}, 1 -> K slots {2,3}
  const int row  = lane & 15;   // q for A rows / s for B cols
  const int waveId = blockIdx.x * (blockDim.x >> 5) + (threadIdx.x >> 5);

  v8f accRe[LMAX + 1];
  v8f accIm[LMAX + 1];
  const v8f vzero = {0.f, 0.f, 0.f, 0.f, 0.f, 0.f, 0.f, 0.f};
#pragma unroll
  for (int p = 0; p <= LMAX; ++p) { accRe[p] = vzero; accIm[p] = vzero; }

  const int ngroups = (N + 3) >> 2;
  for (int g = waveId; g < ngroups; g += NW) {
    // This lane owns two K slots (points): ia = 4g + 2*hi, ib = ia+1.
    const int ia = g * 4 + 2 * hi;
    const int ib = ia + 1;
    float xa = 0.f, ya = 0.f, za = 0.f, xb = 0.f, yb = 0.f, zb = 0.f;
    if (ia < N) { xa = pos[3 * ia]; ya = pos[3 * ia + 1]; za = pos[3 * ia + 2]; }
    if (ib < N) { xb = pos[3 * ib]; yb = pos[3 * ib + 1]; zb = pos[3 * ib + 2]; }

    const float wA = xa * xa + ya * ya + za * za;  // zero for padded points:
    const float wB = xb * xb + yb * yb + zb * zb;  // every A entry carries w
    // xm = 0.5(x - iy), xp = -0.5(x + iy)
    const float xmreA = 0.5f * xa, xmimA = -0.5f * ya;
    const float xmreB = 0.5f * xb, xmimB = -0.5f * yb;
    const float xpreA = -0.5f * xa, xpimA = -0.5f * ya;
    const float xpreB = -0.5f * xb, xpimB = -0.5f * yb;

    // Per-lane exponentiation: base = w * xm^row (complex), bz = z^row.
    float creA = wA, cimA = 0.f, creB = wB, cimB = 0.f;
    float bzA = 1.f, bzB = 1.f;
    for (int j = 0; j < row; ++j) {
      float tA = creA * xmreA - cimA * xmimA;
      cimA = creA * xmimA + cimA * xmreA; creA = tA;
      float tB = creB * xmreB - cimB * xmimB;
      cimB = creB * xmimB + cimB * xmreB; creB = tB;
      bzA *= za; bzB *= zb;
    }
    const v2f Bfrag = {bzA, bzB};  // B(k,s) = z_k^s

    // Running pw = xp^p; 11 p-tiles x {re,im} = 22 WMMAs per 4-point group.
    float pwreA = 1.f, pwimA = 0.f, pwreB = 1.f, pwimB = 0.f;
#pragma unroll
    for (int p = 0; p <= LMAX; ++p) {
      const v2f Are = {creA * pwreA - cimA * pwimA, creB * pwreB - cimB * pwimB};
      const v2f Aim = {creA * pwimA + cimA * pwreA, creB * pwimB + cimB * pwreB};
      accRe[p] = wmma_f32x4(Are, Bfrag, accRe[p]);
      accIm[p] = wmma_f32x4(Aim, Bfrag, accIm[p]);
      float tA = pwreA * xpreA - pwimA * xpimA;
      pwimA = pwreA * xpimA + pwimA * xpreA; pwreA = tA;
      float tB = pwreB * xpreB - pwimB * xpimB;
      pwimB = pwreB * xpimB + pwimB * xpreB; pwreB = tB;
    }
  }

  // Compact 22 C-tiles -> 66 complex partial clm values.
  // C(16x16,f32) layout: element (q,s) at VGPR (q%8), lane s + 16*(q>=8).
  float* dst = part + (size_t)waveId * PART_STRIDE;
#pragma unroll
  for (int l = 0; l <= LMAX; ++l) {
#pragma unroll
    for (int mm = l; mm >= 0; --mm) {
      float sre = 0.f, sim = 0.f;
#pragma unroll
      for (int p = 0; p <= LMAX; ++p) {
        if (p <= l) {
          const int q = p - mm;
          const int s = l - 2 * p + mm;
          if (q >= 0 && s >= 0) {
            const float c = (float)(1.0 / (factd(p) * factd(q) * factd(s)));
            const int srcLane = (q >= 8) ? (s + 16) : s;
            const float re = __shfl(accRe[p][q & 7], srcLane, 32);
            const float im = __shfl(accIm[p][q & 7], srcLane, 32);
            sre += c * re;
            sim += c * im;
          }
        }
      }
      if (lane == 0) {
        const int lm = l * (l + 1) / 2 + (l - mm);
        dst[2 * lm]     = sre;
        dst[2 * lm + 1] = sim;
      }
    }
  }
}

// ---------------------------------------------------------------------------
// Kernel B: sum per-wave partials, apply pref/fac, scatter to 121 outputs.
// ---------------------------------------------------------------------------
__global__ void sht_finalize_kernel(const float* __restrict__ part, int NW,
                                    float* __restrict__ out) {
  const int lm = (int)(blockIdx.x * blockDim.x + threadIdx.x);
  if (lm >= NLM) return;
  // Decode lm -> (l, m): l ascending, m descending l..0.
  int l = 0, base = 0;
  while (base + l + 1 <= lm) { base += l + 1; ++l; }
  const int m = l - (lm - base);

  float sre = 0.f, sim = 0.f;
  for (int w = 0; w < NW; ++w) {
    sre += part[(size_t)w * PART_STRIDE + 2 * lm];
    sim += part[(size_t)w * PART_STRIDE + 2 * lm + 1];
  }

  const double PI_ = 3.14159265358979323846;
  const float pref =
      (float)(sqrt(factd(l + m) * factd(l - m)) * sqrt((2.0 * l + 1.0) / (4.0 * PI_)));
  const float fac = (float)(1.4142135623730951 * ((m & 1) ? -1.0 : 1.0));
  const float cre = pref * sre;
  const float cim = pref * sim;

  const int negI = l * l + l - m;
  const int posI = l * l + l + m;
  out[negI] = (m == 0) ? cre : fac * cim;
  if (m > 0) out[posI] = fac * cre;
}

// ---------------------------------------------------------------------------
extern "C" void kernel_launch(void* const* d_in, const int* in_sizes, int n_in,
                              void* d_out, int out_size, void* d_ws, size_t ws_size,
                              hipStream_t stream) {
  const float* pos = (const float*)d_in[0];
  const int N = in_sizes[0] / 3;  // 500000 points (x,y,z)
  float* out = (float*)d_out;     // 121 f32 outputs
  float* part = (float*)d_ws;

  // One partial record (528 B) per wave; size the grid to the scratch we got.
  int NW = (int)(ws_size / (PART_STRIDE * sizeof(float)));
  NW &= ~7;                 // 8 waves (256 threads) per block
  if (NW > 2048) NW = 2048; // ~61 four-point groups per wave at N=500k
  if (NW < 8) NW = 8;

  sht_partial_kernel<<<dim3(NW / 8), dim3(256), 0, stream>>>(pos, N, part, NW);
  sht_finalize_kernel<<<dim3(1), dim3(128), 0, stream>>>(part, NW, out);
}